// ks_net_block2_16810501996767
// MI455X (gfx1250) — compile-verified
//
#include <hip/hip_runtime.h>

#define Lpix 16384   // 128*128
#define HH   128
#define WW   128

typedef __attribute__((ext_vector_type(2))) float v2f;
typedef __attribute__((ext_vector_type(8))) float v8f;

// ---------------- workspace layout (in floats) ----------------
static const size_t OFF_H    = 0;                       // 16*L  current features, HWC
static const size_t OFF_XF   = 262144;                  // 16*L  conv_first out, NCHW
static const size_t OFF_T    = 524288;                  // 16*L  LN scratch, HWC
static const size_t OFF_XZ   = 786432;                  // 64*L  in_proj out, HWC
static const size_t OFF_XC   = 1835008;                 // 32*L  dwconv+silu, NCHW
static const size_t OFF_XDBL = 2359296;                 // 4*33*L x_proj out [k][c][l]
static const size_t OFF_YS   = 4521984;                 // 4*32*L scan out  [k][d][l]
static const size_t OFF_YSUM = 6619136;                 // 32*L  gated LN'd scan, HWC
static const size_t OFF_YB   = 7143424;                 // 16*L  block residual path, HWC
static const size_t OFF_GIN  = 7405568;                 // 16*L  group residual copy, HWC
// reuse of the XDBL region once scan results are consumed:
static const size_t OFF_C1   = OFF_XDBL;                // 5*L   CAB conv1, NCHW
static const size_t OFF_C2   = OFF_XDBL + 81920;        // 16*L  CAB conv2, NCHW
static const size_t OFF_ATT  = OFF_XDBL + 81920 + 262144; // 64  channel means + attention
static const size_t OFF_RES  = OFF_XDBL;                // 16*L  final conv+xf, NCHW
static const size_t OFF_UP   = OFF_XDBL + 262144;       // 64*L  upsample feats, NCHW

// ---------------- device helpers ----------------
__device__ __forceinline__ float d_sigm(float x){ return 1.f/(1.f+__expf(-x)); }
__device__ __forceinline__ float d_silu(float x){ return x*d_sigm(x); }
__device__ __forceinline__ float d_softplus(float x){ return (x>20.f)? x : log1pf(__expf(x)); }
__device__ __forceinline__ float d_gelu(float x){
  float x3=x*x*x; return 0.5f*x*(1.f+tanhf(0.7978845608f*(x+0.044715f*x3)));
}
// direction mapping into xc[d][*] for the 4 scan orders
__device__ __forceinline__ int xs_map(int k, int l){
  int ll = (k & 2) ? (Lpix-1-l) : l;
  if (k & 1){ int wq = ll >> 7, hq = ll & 127; ll = hq*WW + wq; }
  return ll;
}

// ---------------- TDM (Tensor Data Mover) support ----------------
#if defined(__has_builtin)
#if __has_builtin(__builtin_amdgcn_tensor_load_to_lds) && __has_builtin(__builtin_amdgcn_s_wait_tensorcnt)
#define HAVE_TDM 1
#endif
#endif

#define SCHUNK 64                 // columns per TDM tile (fp32)
#define SROW   66                 // LDS row stride: 64 + 2 pad DWORDs (bank spread)
#define NCHUNK (Lpix/SCHUNK)      // 256 chunks

#ifdef HAVE_TDM
typedef __attribute__((ext_vector_type(4))) unsigned int v4u;
typedef __attribute__((ext_vector_type(4))) int v4i;
typedef __attribute__((ext_vector_type(8))) int v8i;

// Issue a TDM load of a 33-row x 64-col fp32 tile of xdbl[k] (row stride Lpix)
// into LDS at byte offset lds_byte.  D# per CDNA5 ISA ch.8:
//  group0: count=1, lds_addr, global_addr[56:0], type=2
//  group1: data_size=4B, pad_enable, pad_interval=64 DWORDs, pad_amount=2 DWORDs,
//          tensor_dim0=16384, tensor_dim1=33, tile_dim0=64, tile_dim1=33,
//          tensor_dim0_stride=16384
__device__ __forceinline__ void tdm_issue(unsigned lds_byte, const float* gptr)
{
  unsigned long long ga = (unsigned long long)gptr;
  v4u g0;
  g0[0] = 1u;                                              // count=1 (valid)
  g0[1] = lds_byte;                                        // lds_addr
  g0[2] = (unsigned)ga;                                    // global_addr[31:0]
  g0[3] = (unsigned)((ga >> 32) & 0x01FFFFFFu) | (2u<<30); // addr[56:32] | type=2
  v8i g1;
  g1[0] = (int)((2u<<16) | (1u<<20) | (5u<<22) | (1u<<25)); // 4B, pad en, 64dw, +2dw
  g1[1] = (int)(16384u << 16);    // tensor_dim0[15:0] in [31:16]
  g1[2] = (int)(33u << 16);       // tensor_dim0[31:16]=0 | tensor_dim1[15:0]=33
  g1[3] = (int)(64u << 16);       // tensor_dim1[31:16]=0 | tile_dim0=64
  g1[4] = 33;                     // tile_dim1=33 | tile_dim2=0
  g1[5] = 16384;                  // tensor_dim0_stride[31:0]
  g1[6] = 0;                      // stride[47:32]=0 | tensor_dim1_stride lo=0
  g1[7] = 0;
  v4i z4 = {0,0,0,0};
#if __clang_major__ >= 23
  v8i z8 = {0,0,0,0,0,0,0,0};
  __builtin_amdgcn_tensor_load_to_lds(g0, g1, z4, z4, z8, 0);
#else
  __builtin_amdgcn_tensor_load_to_lds(g0, g1, z4, z4, 0);
#endif
}
#endif // HAVE_TDM

// ---------------- kernels ----------------

// conv_first (33->16, 3x3, SAME) + transpose + pe layernorm
__global__ void k_conv_first(const float* __restrict__ x, const float* __restrict__ w,
                             const float* __restrict__ bias, const float* __restrict__ g,
                             const float* __restrict__ beta, float* __restrict__ xf,
                             float* __restrict__ h)
{
  int l = blockIdx.x*blockDim.x + threadIdx.x;
  if (l >= Lpix) return;
  int py = l >> 7, px = l & 127;
  float acc[16];
  #pragma unroll
  for (int o=0;o<16;++o) acc[o] = bias[o];
  for (int c=0;c<33;++c)
    for (int ky=0;ky<3;++ky){
      int iy=py+ky-1; if ((unsigned)iy>=128u) continue;
      for (int kx=0;kx<3;++kx){
        int ix=px+kx-1; if ((unsigned)ix>=128u) continue;
        float xv = x[c*Lpix + iy*WW + ix];
        #pragma unroll
        for (int o=0;o<16;++o) acc[o] += xv * w[o*297 + c*9 + ky*3 + kx];
      }
    }
  float m=0;
  #pragma unroll
  for (int o=0;o<16;++o){ xf[o*Lpix+l]=acc[o]; m+=acc[o]; }
  m *= (1.f/16.f);
  float v=0;
  #pragma unroll
  for (int o=0;o<16;++o){ float d=acc[o]-m; v+=d*d; }
  float r = rsqrtf(v*(1.f/16.f)+1e-5f);
  #pragma unroll
  for (int o=0;o<16;++o) h[l*16+o] = (acc[o]-m)*r*g[o] + beta[o];
}

template<int C>
__global__ void k_layernorm(const float* __restrict__ in, const float* __restrict__ g,
                            const float* __restrict__ b, float* __restrict__ out)
{
  int l = blockIdx.x*blockDim.x + threadIdx.x;
  if (l >= Lpix) return;
  float v[C]; float m=0;
  #pragma unroll
  for (int c=0;c<C;++c){ v[c]=in[l*C+c]; m+=v[c]; }
  m /= (float)C;
  float var=0;
  #pragma unroll
  for (int c=0;c<C;++c){ float d=v[c]-m; var+=d*d; }
  float r = rsqrtf(var/(float)C + 1e-5f);
  #pragma unroll
  for (int c=0;c<C;++c) out[l*C+c] = (v[c]-m)*r*g[c] + b[c];
}

// in_proj: xz(L,64) = t(L,16) @ W(64,16)^T   -- one wave per 16x16 tile, fp32 WMMA
__global__ void k_in_proj(const float* __restrict__ t, const float* __restrict__ w,
                          float* __restrict__ xz)
{
  int m0 = blockIdx.x*16;            // L tile
  int n0 = blockIdx.y*16;            // out-channel tile (0..3)
  int lane = threadIdx.x;
  int mr = lane & 15;
  int koff = (lane >> 4) * 2;
  v8f acc = {};
  #pragma unroll
  for (int kb=0; kb<16; kb+=4){
    v2f a, b;
    a[0] = t[(m0+mr)*16 + kb + koff];
    a[1] = t[(m0+mr)*16 + kb + koff + 1];
    b[0] = w[(n0+mr)*16 + kb + koff];
    b[1] = w[(n0+mr)*16 + kb + koff + 1];
    acc = __builtin_amdgcn_wmma_f32_16x16x4_f32(false, a, false, b, (short)0, acc, false, false);
  }
  #pragma unroll
  for (int r=0;r<8;++r){
    int m = r + ((lane>>4)<<3);
    xz[(m0+m)*64 + n0 + mr] = acc[r];
  }
}

// depthwise 3x3 conv on xa (cols 0..31 of xz) + bias + SiLU -> xc NCHW
__global__ void k_dwconv_silu(const float* __restrict__ xz, const float* __restrict__ w,
                              const float* __restrict__ b, float* __restrict__ xc)
{
  int i = blockIdx.x*blockDim.x + threadIdx.x;
  if (i >= 32*Lpix) return;
  int d = i >> 14, l = i & (Lpix-1);
  int py=l>>7, px=l&127;
  float acc = b[d];
  #pragma unroll
  for (int ky=0;ky<3;++ky){
    int iy=py+ky-1; if ((unsigned)iy>=128u) continue;
    #pragma unroll
    for (int kx=0;kx<3;++kx){
      int ix=px+kx-1; if ((unsigned)ix>=128u) continue;
      acc += xz[(iy*WW+ix)*64 + d] * w[d*9 + ky*3 + kx];
    }
  }
  xc[d*Lpix + l] = d_silu(acc);
}

// x_proj: xdbl[k][c][l] = sum_d xpw[k][c][d] * xs[k][d][l], fp32 WMMA, M=33 padded to 48
__global__ void k_x_proj(const float* __restrict__ xc, const float* __restrict__ xpw,
                         float* __restrict__ xdbl)
{
  int n0 = blockIdx.x*16;            // l tile
  int mt = blockIdx.y;               // c tile 0..2
  int k  = blockIdx.z;               // direction
  int lane = threadIdx.x;
  int mr = lane & 15;
  int koff = (lane >> 4) * 2;
  const float* wk = xpw + k*33*32;
  int m = mt*16 + mr;
  int mclamp = (m < 33) ? m : 32;
  float amask = (m < 33) ? 1.f : 0.f;
  int col = xs_map(k, n0 + mr);
  v8f acc = {};
  #pragma unroll
  for (int kb=0; kb<32; kb+=4){
    v2f a, b;
    a[0] = amask * wk[mclamp*32 + kb + koff];
    a[1] = amask * wk[mclamp*32 + kb + koff + 1];
    b[0] = xc[(kb+koff  )*Lpix + col];
    b[1] = xc[(kb+koff+1)*Lpix + col];
    acc = __builtin_amdgcn_wmma_f32_16x16x4_f32(false, a, false, b, (short)0, acc, false, false);
  }
  #pragma unroll
  for (int r=0;r<8;++r){
    int mm = mt*16 + r + ((lane>>4)<<3);
    if (mm < 33) xdbl[(k*33+mm)*Lpix + n0 + mr] = acc[r];
  }
}

// selective scan: 1 block per direction, 512 threads = (d,n) state grid.
// TDM double-buffers 33x64 column chunks of xdbl[k] into LDS while the serial
// recurrence runs; shfl_xor reduces over n within each 16-lane group.
__global__ void k_scan(const float* __restrict__ xdbl, const float* __restrict__ xc,
                       const float* __restrict__ A_log, const float* __restrict__ dtw,
                       const float* __restrict__ dtb, const float* __restrict__ Ds,
                       float* __restrict__ y)
{
  __shared__ float sb[2][33*SROW];    // only static LDS object -> offset 0
  int k = blockIdx.x;
  int t = threadIdx.x;               // 0..511
  int d = t >> 4, n = t & 15;
  float A   = -__expf(A_log[(k*32+d)*16 + n]);
  float wdt = dtw[k*32+d];
  float bdt = dtb[k*32+d];
  float Dv  = Ds[k*32+d];
  const float* gbase = xdbl + (size_t)k*33*Lpix;
  const float* u_row = xc + (size_t)d*Lpix;
  float* y_row = y + (size_t)(k*32 + d)*Lpix;
  float h = 0.f;
#ifdef HAVE_TDM
  const unsigned ldsBytes = (unsigned)(33*SROW*sizeof(float));
  bool leader = (t < 32);            // wave 0 drives the TDM pipeline
  if (leader) tdm_issue(0u, gbase);  // chunk 0 -> buffer 0
  for (int c=0; c<NCHUNK; ++c){
    if (leader) __builtin_amdgcn_s_wait_tensorcnt(0);
    __syncthreads();                 // chunk c resident for all waves
    asm volatile("" ::: "memory");
    if (leader && (c+1) < NCHUNK)
      tdm_issue(((c+1)&1) ? ldsBytes : 0u, gbase + (size_t)(c+1)*SCHUNK);
    const float* cb = sb[c & 1];
    int l0 = c*SCHUNK;
    for (int j=0; j<SCHUNK; ++j){
      int l = l0 + j;
      float delta = d_softplus(wdt*cb[j] + bdt);          // dt row = row 0
      float u  = u_row[xs_map(k, l)];
      float Bv = cb[(1+n)*SROW + j];
      float Cv = cb[(17+n)*SROW + j];
      h = h*__expf(delta*A) + delta*u*Bv;
      float p = h*Cv;
      p += __shfl_xor(p, 8, 32);
      p += __shfl_xor(p, 4, 32);
      p += __shfl_xor(p, 2, 32);
      p += __shfl_xor(p, 1, 32);
      if (n == 0) y_row[l] = p + Dv*u;
    }
    __syncthreads();                 // everyone done reading before reuse
  }
#else
  const float* dt_row = gbase;
  const float* B_row  = gbase + (size_t)(1 + n)*Lpix;
  const float* C_row  = gbase + (size_t)(17 + n)*Lpix;
  for (int l=0; l<Lpix; ++l){
    if ((l & 63) == 0){
      __builtin_prefetch(B_row + l + 64, 0, 0);
      __builtin_prefetch(C_row + l + 64, 0, 0);
    }
    float delta = d_softplus(wdt * dt_row[l] + bdt);
    float u  = u_row[xs_map(k, l)];
    float Bv = B_row[l], Cv = C_row[l];
    h = h*__expf(delta*A) + delta*u*Bv;
    float p = h*Cv;
    p += __shfl_xor(p, 8, 32);
    p += __shfl_xor(p, 4, 32);
    p += __shfl_xor(p, 2, 32);
    p += __shfl_xor(p, 1, 32);
    if (n == 0) y_row[l] = p + Dv*u;
  }
  (void)sb;
#endif
}

// combine 4 directions (reverse/transpose) + out_norm LN + silu(z) gating -> (L,32)
__global__ void k_combine(const float* __restrict__ y, const float* __restrict__ xz,
                          const float* __restrict__ g, const float* __restrict__ b,
                          float* __restrict__ out)
{
  int l = blockIdx.x*blockDim.x + threadIdx.x;
  if (l >= Lpix) return;
  int hq = l >> 7, wq = l & 127;
  int lt = wq*WW + hq;
  float v[32]; float m=0;
  #pragma unroll
  for (int d=0; d<32; ++d){
    float s = y[(0*32+d)*Lpix + l]
            + y[(2*32+d)*Lpix + (Lpix-1-l)]
            + y[(1*32+d)*Lpix + lt]
            + y[(3*32+d)*Lpix + (Lpix-1-lt)];
    v[d]=s; m+=s;
  }
  m *= (1.f/32.f);
  float var=0;
  #pragma unroll
  for (int d=0; d<32; ++d){ float q=v[d]-m; var+=q*q; }
  float r = rsqrtf(var*(1.f/32.f) + 1e-5f);
  #pragma unroll
  for (int d=0; d<32; ++d)
    out[l*32+d] = ((v[d]-m)*r*g[d] + b[d]) * d_silu(xz[l*64 + 32 + d]);
}

// out_proj (32->16) fp32 WMMA + skip1 residual
__global__ void k_out_proj(const float* __restrict__ gin, const float* __restrict__ opw,
                           const float* __restrict__ h, const float* __restrict__ skip1,
                           float* __restrict__ yb)
{
  int m0 = blockIdx.x*16;
  int lane = threadIdx.x;
  int mr = lane & 15;
  int koff = (lane >> 4) * 2;
  v8f acc = {};
  #pragma unroll
  for (int kb=0; kb<32; kb+=4){
    v2f a, b;
    a[0] = gin[(m0+mr)*32 + kb + koff];
    a[1] = gin[(m0+mr)*32 + kb + koff + 1];
    b[0] = opw[mr*32 + kb + koff];
    b[1] = opw[mr*32 + kb + koff + 1];
    acc = __builtin_amdgcn_wmma_f32_16x16x4_f32(false, a, false, b, (short)0, acc, false, false);
  }
  #pragma unroll
  for (int r=0;r<8;++r){
    int m = m0 + r + ((lane>>4)<<3);
    yb[m*16 + mr] = acc[r] + h[m*16 + mr]*skip1[mr];
  }
}

// CAB conv1: HWC(16) -> NCHW(5), 3x3, gelu
__global__ void k_cab_c1(const float* __restrict__ t, const float* __restrict__ w,
                         const float* __restrict__ b, float* __restrict__ o)
{
  int i = blockIdx.x*blockDim.x + threadIdx.x;
  if (i >= 5*Lpix) return;
  int oc = i >> 14, l = i & (Lpix-1);
  int py=l>>7, px=l&127;
  float acc = b[oc];
  for (int ky=0;ky<3;++ky){
    int iy=py+ky-1; if ((unsigned)iy>=128u) continue;
    for (int kx=0;kx<3;++kx){
      int ix=px+kx-1; if ((unsigned)ix>=128u) continue;
      const float* tp = t + (iy*WW+ix)*16;
      const float* wp = w + oc*144 + ky*3 + kx;
      #pragma unroll
      for (int c=0;c<16;++c) acc += tp[c]*wp[c*9];
    }
  }
  o[oc*Lpix + l] = d_gelu(acc);
}

// CAB conv2: NCHW(5) -> NCHW(16), 3x3
__global__ void k_cab_c2(const float* __restrict__ c1, const float* __restrict__ w,
                         const float* __restrict__ b, float* __restrict__ o)
{
  int i = blockIdx.x*blockDim.x + threadIdx.x;
  if (i >= 16*Lpix) return;
  int oc = i >> 14, l = i & (Lpix-1);
  int py=l>>7, px=l&127;
  float acc = b[oc];
  for (int ky=0;ky<3;++ky){
    int iy=py+ky-1; if ((unsigned)iy>=128u) continue;
    for (int kx=0;kx<3;++kx){
      int ix=px+kx-1; if ((unsigned)ix>=128u) continue;
      #pragma unroll
      for (int c=0;c<5;++c)
        acc += c1[c*Lpix + iy*WW + ix] * w[oc*45 + c*9 + ky*3 + kx];
    }
  }
  o[oc*Lpix + l] = acc;
}

__global__ void k_chan_mean(const float* __restrict__ c2, float* __restrict__ mean)
{
  __shared__ float red[256];
  int c = blockIdx.x, t = threadIdx.x;
  float s = 0;
  for (int l=t; l<Lpix; l+=256) s += c2[c*Lpix + l];
  red[t] = s; __syncthreads();
  for (int o=128; o>0; o>>=1){ if (t<o) red[t]+=red[t+o]; __syncthreads(); }
  if (t==0) mean[c] = red[0]*(1.f/(float)Lpix);
}

__global__ void k_chan_att(const float* __restrict__ mean, const float* __restrict__ a1w,
                           const float* __restrict__ a1b, const float* __restrict__ a2w,
                           const float* __restrict__ a2b, float* __restrict__ att)
{
  int c = threadIdx.x; if (c >= 16) return;
  float s = a1b[0];
  #pragma unroll
  for (int j=0;j<16;++j) s += mean[j]*a1w[j];
  s = s > 0.f ? s : 0.f;
  att[c] = d_sigm(s*a2w[c] + a2b[c]);
}

// y*skip2 + transpose(cab * att) -> next h (HWC)
__global__ void k_block_out(const float* __restrict__ yb, const float* __restrict__ c2,
                            const float* __restrict__ att, const float* __restrict__ skip2,
                            float* __restrict__ h)
{
  int i = blockIdx.x*blockDim.x + threadIdx.x;
  if (i >= 16*Lpix) return;
  int l = i >> 4, c = i & 15;
  h[l*16+c] = yb[l*16+c]*skip2[c] + c2[c*Lpix + l]*att[c];
}

// group conv (HWC in, 16->16, 3x3) + group residual, HWC out
__global__ void k_group_conv(const float* __restrict__ hin, const float* __restrict__ w,
                             const float* __restrict__ b, const float* __restrict__ gin,
                             float* __restrict__ hout)
{
  int i = blockIdx.x*blockDim.x + threadIdx.x;
  if (i >= 16*Lpix) return;
  int oc = i & 15, l = i >> 4;
  int py=l>>7, px=l&127;
  float acc = b[oc];
  for (int ky=0;ky<3;++ky){
    int iy=py+ky-1; if ((unsigned)iy>=128u) continue;
    for (int kx=0;kx<3;++kx){
      int ix=px+kx-1; if ((unsigned)ix>=128u) continue;
      const float* hp = hin + (iy*WW+ix)*16;
      const float* wp = w + oc*144 + ky*3 + kx;
      #pragma unroll
      for (int c=0;c<16;++c) acc += hp[c]*wp[c*9];
    }
  }
  hout[l*16+oc] = acc + gin[l*16+oc];
}

// final cab conv (HWC in, 16->16) + xf residual, NCHW out
__global__ void k_final_cab(const float* __restrict__ t, const float* __restrict__ w,
                            const float* __restrict__ b, const float* __restrict__ xf,
                            float* __restrict__ res)
{
  int i = blockIdx.x*blockDim.x + threadIdx.x;
  if (i >= 16*Lpix) return;
  int oc = i >> 14, l = i & (Lpix-1);
  int py=l>>7, px=l&127;
  float acc = b[oc];
  for (int ky=0;ky<3;++ky){
    int iy=py+ky-1; if ((unsigned)iy>=128u) continue;
    for (int kx=0;kx<3;++kx){
      int ix=px+kx-1; if ((unsigned)ix>=128u) continue;
      const float* tp = t + (iy*WW+ix)*16;
      const float* wp = w + oc*144 + ky*3 + kx;
      #pragma unroll
      for (int c=0;c<16;++c) acc += tp[c]*wp[c*9];
    }
  }
  res[oc*Lpix + l] = acc + xf[oc*Lpix + l];
}

// up conv (NCHW 16->64, 3x3) + leaky relu 0.01
__global__ void k_up(const float* __restrict__ res, const float* __restrict__ w,
                     const float* __restrict__ b, float* __restrict__ up)
{
  int i = blockIdx.x*blockDim.x + threadIdx.x;
  if (i >= 64*Lpix) return;
  int oc = i >> 14, l = i & (Lpix-1);
  int py=l>>7, px=l&127;
  float acc = b[oc];
  for (int ky=0;ky<3;++ky){
    int iy=py+ky-1; if ((unsigned)iy>=128u) continue;
    for (int kx=0;kx<3;++kx){
      int ix=px+kx-1; if ((unsigned)ix>=128u) continue;
      #pragma unroll
      for (int c=0;c<16;++c)
        acc += res[c*Lpix + iy*WW + ix] * w[oc*144 + c*9 + ky*3 + kx];
    }
  }
  up[oc*Lpix + l] = acc > 0.f ? acc : 0.01f*acc;
}

// last conv (NCHW 64->33, 3x3) + abs -> output
__global__ void k_last(const float* __restrict__ up, const float* __restrict__ w,
                       const float* __restrict__ b, float* __restrict__ out)
{
  int i = blockIdx.x*blockDim.x + threadIdx.x;
  if (i >= 33*Lpix) return;
  int oc = i >> 14, l = i & (Lpix-1);
  int py=l>>7, px=l&127;
  float acc = b[oc];
  for (int ky=0;ky<3;++ky){
    int iy=py+ky-1; if ((unsigned)iy>=128u) continue;
    for (int kx=0;kx<3;++kx){
      int ix=px+kx-1; if ((unsigned)ix>=128u) continue;
      for (int c=0;c<64;++c)
        acc += up[c*Lpix + iy*WW + ix] * w[oc*576 + c*9 + ky*3 + kx];
    }
  }
  out[oc*Lpix + l] = fabsf(acc);
}

// ---------------- host side ----------------
// Param flatten order (jax tree: dict keys sorted, uppercase < lowercase), x at index 0:
// 1 cab_b, 2 cab_w, 3 conv_first_b, 4 conv_first_w,
// 5..29 group0.block0 (A_log,Ds,a1_b,a1_w,a2_b,a2_w,c1_b,c1_w,c2_b,c2_w,conv_b,conv_w,
//        dt_b,dt_w,in_proj_w,ln1_b,ln1_g,ln2_b,ln2_g,out_norm_b,out_norm_g,out_proj_w,
//        skip1,skip2,x_proj_w),
// 30 group0.conv_b, 31 group0.conv_w,
// 32..56 group1.block0, 57..81 group1.block1, 82 group1.conv_b, 83 group1.conv_w,
// 84 last_b, 85 last_w, 86 norm_b, 87 norm_g, 88 pe_norm_b, 89 pe_norm_g, 90 up_b, 91 up_w

static void run_vss_block(void* const* d_in, float* ws, int B, hipStream_t stream)
{
  auto P = [&](int i){ return (const float*)d_in[i]; };
  k_layernorm<16><<<Lpix/256, 256, 0, stream>>>(ws+OFF_H, P(B+16), P(B+15), ws+OFF_T);
  dim3 gip(Lpix/16, 4);
  k_in_proj<<<gip, 32, 0, stream>>>(ws+OFF_T, P(B+14), ws+OFF_XZ);
  k_dwconv_silu<<<32*Lpix/256, 256, 0, stream>>>(ws+OFF_XZ, P(B+11), P(B+10), ws+OFF_XC);
  dim3 gxp(Lpix/16, 3, 4);
  k_x_proj<<<gxp, 32, 0, stream>>>(ws+OFF_XC, P(B+24), ws+OFF_XDBL);
  k_scan<<<4, 512, 0, stream>>>(ws+OFF_XDBL, ws+OFF_XC, P(B+0), P(B+13), P(B+12), P(B+1), ws+OFF_YS);
  k_combine<<<Lpix/256, 256, 0, stream>>>(ws+OFF_YS, ws+OFF_XZ, P(B+20), P(B+19), ws+OFF_YSUM);
  k_out_proj<<<Lpix/16, 32, 0, stream>>>(ws+OFF_YSUM, P(B+21), ws+OFF_H, P(B+22), ws+OFF_YB);
  k_layernorm<16><<<Lpix/256, 256, 0, stream>>>(ws+OFF_YB, P(B+18), P(B+17), ws+OFF_T);
  k_cab_c1<<<5*Lpix/256, 256, 0, stream>>>(ws+OFF_T, P(B+7), P(B+6), ws+OFF_C1);
  k_cab_c2<<<16*Lpix/256, 256, 0, stream>>>(ws+OFF_C1, P(B+9), P(B+8), ws+OFF_C2);
  k_chan_mean<<<16, 256, 0, stream>>>(ws+OFF_C2, ws+OFF_ATT);
  k_chan_att<<<1, 16, 0, stream>>>(ws+OFF_ATT, P(B+3), P(B+2), P(B+5), P(B+4), ws+OFF_ATT+16);
  k_block_out<<<16*Lpix/256, 256, 0, stream>>>(ws+OFF_YB, ws+OFF_C2, ws+OFF_ATT+16, P(B+23), ws+OFF_H);
}

extern "C" void kernel_launch(void* const* d_in, const int* in_sizes, int n_in,
                              void* d_out, int out_size, void* d_ws, size_t ws_size,
                              hipStream_t stream)
{
  (void)in_sizes; (void)n_in; (void)out_size; (void)ws_size;
  const float* x = (const float*)d_in[0];
  auto P = [&](int i){ return (const float*)d_in[i]; };
  float* ws = (float*)d_ws;
  float* out = (float*)d_out;

  // conv_first + pe layernorm
  k_conv_first<<<Lpix/256, 256, 0, stream>>>(x, P(4), P(3), P(89), P(88),
                                             ws+OFF_XF, ws+OFF_H);

  const int blockBase[2][2] = {{5, -1}, {32, 57}};
  const int gconvB[2] = {30, 82};   // group conv bias index; weight = +1
  const int depths[2] = {1, 2};
  for (int gi=0; gi<2; ++gi){
    hipMemcpyAsync(ws+OFF_GIN, ws+OFF_H, (size_t)16*Lpix*sizeof(float),
                   hipMemcpyDeviceToDevice, stream);
    for (int bi=0; bi<depths[gi]; ++bi)
      run_vss_block(d_in, ws, blockBase[gi][bi], stream);
    k_group_conv<<<16*Lpix/256, 256, 0, stream>>>(ws+OFF_H, P(gconvB[gi]+1), P(gconvB[gi]),
                                                  ws+OFF_GIN, ws+OFF_T);
    hipMemcpyAsync(ws+OFF_H, ws+OFF_T, (size_t)16*Lpix*sizeof(float),
                   hipMemcpyDeviceToDevice, stream);
  }

  // final norm, cab conv + xf residual, upsample-feat conv + lrelu, last conv + abs
  k_layernorm<16><<<Lpix/256, 256, 0, stream>>>(ws+OFF_H, P(87), P(86), ws+OFF_T);
  k_final_cab<<<16*Lpix/256, 256, 0, stream>>>(ws+OFF_T, P(2), P(1), ws+OFF_XF, ws+OFF_RES);
  k_up<<<64*Lpix/256, 256, 0, stream>>>(ws+OFF_RES, P(91), P(90), ws+OFF_UP);
  k_last<<<33*Lpix/256, 256, 0, stream>>>(ws+OFF_UP, P(85), P(84), out);
}